// GCN_14577119003090
// MI455X (gfx1250) — compile-verified
//
#include <hip/hip_runtime.h>
#include <math.h>

typedef __attribute__((ext_vector_type(2))) float v2f;
typedef __attribute__((ext_vector_type(8))) float v8f;

// ---------------------------------------------------------------------------
// Workspace layout (floats), n = N_NODES, g = N_GRAPHS:
//   [0, n)              deg  -> dinv (in place)
//   [n, 65n)            h1   [n,64]   (fully written by WMMA kernel)
//   [65n, 129n)         agg1 [n,64]   (zeroed)
//   [129n, 132n)        agg2 [n,3]    (zeroed)
//   [132n, 132n+3g)     pooled [g,3]  (zeroed)
//   [132n+3g, 132n+4g)  cnt  [g]      (zeroed)
//   [132n+4g, 135n+4g)  h2   [n,3]    (fully written)
// Total = 135n + 4g floats  (~103 MB for n=200000, g=512)
// ---------------------------------------------------------------------------

__global__ void gcn_init_kernel(float* __restrict__ ws, int n, int g) {
  const long long zbase  = 65LL * n;            // start of zeroed region
  const long long zcount = 67LL * n + 4LL * g;  // agg1+agg2+pooled+cnt
  const long long total  = (long long)n + zcount;
  const long long stride = (long long)gridDim.x * blockDim.x;
  for (long long i = blockIdx.x * (long long)blockDim.x + threadIdx.x;
       i < total; i += stride) {
    if (i < n) ws[i] = 1.0f;                    // self-loop contribution to deg
    else       ws[zbase + (i - (long long)n)] = 0.0f;
  }
}

__global__ void gcn_deg_kernel(const int* __restrict__ dst,
                               float* __restrict__ deg, int e) {
  int i = blockIdx.x * blockDim.x + threadIdx.x;
  if (i < e) atomicAdd(&deg[dst[i]], 1.0f);
}

__global__ void gcn_dinv_kernel(float* __restrict__ deg, int n) {
  int i = blockIdx.x * blockDim.x + threadIdx.x;
  if (i < n) deg[i] = rsqrtf(fmaxf(deg[i], 1.0f));
}

// h1[n,64] = x[n,2] @ W1[2,64] using V_WMMA_F32_16X16X4_F32 (K padded 2->4).
// One wave32 processes TPW 16-node tiles; 4 WMMAs cover the 64 hidden dims.
// B (W1) tiles are loaded once per wave and reused across all node tiles.
// A layout (ISA 7.12.2): V0 = K0(lanes0-15)/K2(lanes16-31), V1 = K1/K3.
// B layout mirrors A with lanes encoding N. K=2,3 slots are the zero pad.
// C/D: VGPR r -> row r (lanes 0-15, col=lane) / row r+8 (lanes 16-31).
#define TPW 8  // 16-node tiles per wave
__global__ void gcn_h1_wmma_kernel(const float* __restrict__ x,
                                   const float* __restrict__ W1,
                                   float* __restrict__ h1, int n) {
  const int  lane = threadIdx.x & 31;
  const int  wave = (int)((blockIdx.x * (unsigned)blockDim.x + threadIdx.x) >> 5);
  const int  m    = lane & 15;
  const bool lo   = lane < 16;

  // Load all 4 B-tiles of W1 (2x64) once; hi-lane K=2,3 slots are zero pad.
  v2f B[4];
#pragma unroll
  for (int t = 0; t < 4; ++t) {
    B[t][0] = lo ? W1[0 * 64 + t * 16 + m] : 0.0f;   // W1 row K=0
    B[t][1] = lo ? W1[1 * 64 + t * 16 + m] : 0.0f;   // W1 row K=1
  }

  const int tiles   = (n + 15) >> 4;
  const int tile_lo = wave * TPW;
  const int tile_hi = (tile_lo + TPW < tiles) ? (tile_lo + TPW) : tiles;

  for (int tile = tile_lo; tile < tile_hi; ++tile) {   // wave-uniform bounds
    const int  node0 = tile << 4;
    const int  node  = node0 + m;
    const bool full  = (node0 + 16 <= n);              // wave-uniform

    const bool okA = lo && (node < n);
    v2f A;
    A[0] = okA ? x[2 * node + 0] : 0.0f;               // K=0 (lo) / K=2 pad
    A[1] = okA ? x[2 * node + 1] : 0.0f;               // K=1 (lo) / K=3 pad

    // Per-lane output base: row block (lo: rows 0-7, hi: rows 8-15), col m.
    float* outp = h1 + (long long)node0 * 64 + (lo ? 0 : 8 * 64) + m;

    if (full) {
#pragma unroll
      for (int t = 0; t < 4; ++t) {
        v8f C = {};
        v8f D = __builtin_amdgcn_wmma_f32_16x16x4_f32(
            false, A, false, B[t], (short)0, C, false, false);
#pragma unroll
        for (int r = 0; r < 8; ++r)
          outp[r * 64 + t * 16] = D[r];                // immediate offsets
      }
    } else {
      const int rbase = lo ? 0 : 8;
#pragma unroll
      for (int t = 0; t < 4; ++t) {
        v8f C = {};
        v8f D = __builtin_amdgcn_wmma_f32_16x16x4_f32(
            false, A, false, B[t], (short)0, C, false, false);
#pragma unroll
        for (int r = 0; r < 8; ++r)
          if (node0 + rbase + r < n) outp[r * 64 + t * 16] = D[r];
      }
    }
  }
}

// Layer-1 aggregation: 16 lanes per (edge|self-loop); float4 gather of
// h1[src] (256B contiguous per edge across the 16 lanes) + 4 atomic adds.
__global__ void gcn_agg1_kernel(const int* __restrict__ src,
                                const int* __restrict__ dst,
                                const float* __restrict__ dinv,
                                const float* __restrict__ h1,
                                float* __restrict__ agg1, int e, int n) {
  const long long tid   = blockIdx.x * (long long)blockDim.x + threadIdx.x;
  const long long idx   = tid >> 4;             // edge id (incl. self loops)
  const int       t     = (int)(tid & 15);      // 4-dim group
  const long long total = (long long)e + n;
  if (idx >= total) return;
  int u, v;
  if (idx < e) { u = src[idx]; v = dst[idx]; }
  else         { u = v = (int)(idx - (long long)e); }
  const float w = dinv[u] * dinv[v];
  const float4 hv = *reinterpret_cast<const float4*>(h1 + (long long)u * 64 + t * 4);
  float* o = agg1 + (long long)v * 64 + t * 4;
  atomicAdd(o + 0, hv.x * w);
  atomicAdd(o + 1, hv.y * w);
  atomicAdd(o + 2, hv.z * w);
  atomicAdd(o + 3, hv.w * w);
}

// Fused relu(agg1 + b1) @ W2 -> h2[n,3].  Avoids materializing hrelu
// (~100 MB of saved L2 round trips).  b1/W2 are uniform -> scalar loads.
__global__ void gcn_h2_kernel(const float* __restrict__ agg1,
                              const float* __restrict__ b1,
                              const float* __restrict__ W2,
                              float* __restrict__ h2, int n) {
  int i = blockIdx.x * blockDim.x + threadIdx.x;
  if (i >= n) return;
  const float* a = agg1 + (long long)i * 64;
  float a0 = 0.0f, a1 = 0.0f, a2 = 0.0f;
#pragma unroll
  for (int d = 0; d < 64; ++d) {
    const float h = fmaxf(a[d] + b1[d], 0.0f);
    a0 = fmaf(h, W2[d * 3 + 0], a0);
    a1 = fmaf(h, W2[d * 3 + 1], a1);
    a2 = fmaf(h, W2[d * 3 + 2], a2);
  }
  h2[3LL * i + 0] = a0;
  h2[3LL * i + 1] = a1;
  h2[3LL * i + 2] = a2;
}

__global__ void gcn_agg2_kernel(const int* __restrict__ src,
                                const int* __restrict__ dst,
                                const float* __restrict__ dinv,
                                const float* __restrict__ h2,
                                float* __restrict__ agg2, int e, int n) {
  const long long idx   = blockIdx.x * (long long)blockDim.x + threadIdx.x;
  const long long total = (long long)e + n;
  if (idx >= total) return;
  int u, v;
  if (idx < e) { u = src[idx]; v = dst[idx]; }
  else         { u = v = (int)(idx - (long long)e); }
  const float w = dinv[u] * dinv[v];
  const float* h = h2 + (long long)u * 3;
  float* o = agg2 + (long long)v * 3;
  atomicAdd(o + 0, h[0] * w);
  atomicAdd(o + 1, h[1] * w);
  atomicAdd(o + 2, h[2] * w);
}

__global__ void gcn_pool_kernel(const float* __restrict__ agg2,
                                const float* __restrict__ b2,
                                const int* __restrict__ batch,
                                float* __restrict__ pooled,
                                float* __restrict__ cnt, int n) {
  int i = blockIdx.x * blockDim.x + threadIdx.x;
  if (i >= n) return;
  const int gidx = batch[i];
  atomicAdd(&pooled[gidx * 3 + 0], agg2[3LL * i + 0] + b2[0]);
  atomicAdd(&pooled[gidx * 3 + 1], agg2[3LL * i + 1] + b2[1]);
  atomicAdd(&pooled[gidx * 3 + 2], agg2[3LL * i + 2] + b2[2]);
  atomicAdd(&cnt[gidx], 1.0f);
}

__global__ void gcn_logsoftmax_kernel(const float* __restrict__ pooled,
                                      const float* __restrict__ cnt,
                                      float* __restrict__ out, int g) {
  int i = blockIdx.x * blockDim.x + threadIdx.x;
  if (i >= g) return;
  const float c  = fmaxf(cnt[i], 1.0f);
  const float p0 = pooled[i * 3 + 0] / c;
  const float p1 = pooled[i * 3 + 1] / c;
  const float p2 = pooled[i * 3 + 2] / c;
  const float mx  = fmaxf(p0, fmaxf(p1, p2));
  const float s   = expf(p0 - mx) + expf(p1 - mx) + expf(p2 - mx);
  const float lse = mx + logf(s);
  out[i * 3 + 0] = p0 - lse;
  out[i * 3 + 1] = p1 - lse;
  out[i * 3 + 2] = p2 - lse;
}

extern "C" void kernel_launch(void* const* d_in, const int* in_sizes, int n_in,
                              void* d_out, int out_size, void* d_ws, size_t ws_size,
                              hipStream_t stream) {
  const float* x     = (const float*)d_in[0];
  const int*   ei    = (const int*)  d_in[1];
  const int*   batch = (const int*)  d_in[2];
  const float* W1    = (const float*)d_in[3];
  const float* b1    = (const float*)d_in[4];
  const float* W2    = (const float*)d_in[5];
  const float* b2    = (const float*)d_in[6];
  float* out = (float*)d_out;

  const int n = in_sizes[0] / 2;   // D_IN  = 2
  const int e = in_sizes[1] / 2;   // edge_index is [2, E]
  const int g = out_size / 3;      // D_OUT = 3

  const int* src = ei;             // row 0
  const int* dst = ei + e;         // row 1

  float* ws     = (float*)d_ws;
  float* deg    = ws;                          // n (becomes dinv in place)
  float* h1     = ws + (long long)n;           // 64n
  float* agg1   = ws + 65LL * n;               // 64n
  float* agg2   = ws + 129LL * n;              // 3n
  float* pooled = ws + 132LL * n;              // 3g
  float* cnt    = pooled + 3LL * g;            // g
  float* h2     = cnt + g;                     // 3n

  // 1) init workspace
  gcn_init_kernel<<<2048, 256, 0, stream>>>(ws, n, g);
  // 2) degree (self loop already counted via init = 1.0)
  gcn_deg_kernel<<<(e + 255) / 256, 256, 0, stream>>>(dst, deg, e);
  // 3) dinv in place
  gcn_dinv_kernel<<<(n + 255) / 256, 256, 0, stream>>>(deg, n);
  // 4) h1 = x @ W1 via WMMA f32 16x16x4 (TPW tiles per wave)
  {
    const int tiles = (n + 15) / 16;
    const int waves = (tiles + TPW - 1) / TPW;
    const long long thr = (long long)waves * 32;
    gcn_h1_wmma_kernel<<<(int)((thr + 255) / 256), 256, 0, stream>>>(x, W1, h1, n);
  }
  // 5) layer-1 aggregation (16 lanes per edge)
  {
    const long long work = ((long long)e + n) * 16;
    gcn_agg1_kernel<<<(int)((work + 255) / 256), 256, 0, stream>>>(src, dst, deg, h1, agg1, e, n);
  }
  // 6) fused relu + layer-2 GEMM
  gcn_h2_kernel<<<(n + 255) / 256, 256, 0, stream>>>(agg1, b1, W2, h2, n);
  // 7) layer-2 aggregation
  {
    const long long work = (long long)e + n;
    gcn_agg2_kernel<<<(int)((work + 255) / 256), 256, 0, stream>>>(src, dst, deg, h2, agg2, e, n);
  }
  // 8) mean pool (sum + count)
  gcn_pool_kernel<<<(n + 255) / 256, 256, 0, stream>>>(agg2, b2, batch, pooled, cnt, n);
  // 9) log-softmax
  gcn_logsoftmax_kernel<<<(g + 255) / 256, 256, 0, stream>>>(pooled, cnt, out, g);
}